// HashGridEncoder_14903536517280
// MI455X (gfx1250) — compile-verified
//
#include <hip/hip_runtime.h>
#include <cmath>

#define LVLS   16
#define LOG2T  19
#define HMASK  ((1u << LOG2T) - 1u)

typedef float vf2 __attribute__((ext_vector_type(2)));

struct Res16 { float v[LVLS]; };

__global__ __launch_bounds__(256) void hashgrid_fwd(
    const float* __restrict__ x, const float* __restrict__ table,
    float* __restrict__ out, Res16 res, int npts)
{
    __shared__ __align__(16) float sx[256 * 3];
    const int tid  = threadIdx.x;
    const int base = blockIdx.x * 256;
    const int n    = base + tid;
    const bool full = (base + 256 <= npts);

    if (full) {
        // Stage this block's 3KB slice of x into LDS with the CDNA5 async
        // copy engine: 192 lanes x 16B = 3072B, fully coalesced, ASYNCcnt-tracked.
        if (tid < 192) {
            unsigned long long ga =
                (unsigned long long)(x + (size_t)base * 3 + (size_t)tid * 4);
            unsigned int loff = (unsigned int)(uintptr_t)(&sx[tid * 4]);
            asm volatile("global_load_async_to_lds_b128 %0, %1, off"
                         :: "v"(loff), "v"(ga) : "memory");
        }
        asm volatile("s_wait_asynccnt 0" ::: "memory");
        __syncthreads();
    } else {
        if (n < npts) {
            sx[tid * 3 + 0] = x[(size_t)n * 3 + 0];
            sx[tid * 3 + 1] = x[(size_t)n * 3 + 1];
            sx[tid * 3 + 2] = x[(size_t)n * 3 + 2];
        }
        __syncthreads();
    }
    if (n >= npts) return;

    const float px = sx[tid * 3 + 0];
    const float py = sx[tid * 3 + 1];
    const float pz = sx[tid * 3 + 2];
    float* orow = out + ((size_t)n << 5);   // 32 floats per point = 1 cacheline

    #pragma unroll
    for (int lvl = 0; lvl < LVLS; ++lvl) {
        const float r = res.v[lvl];
        const float xs0 = px * r, xs1 = py * r, xs2 = pz * r;
        const float f0 = floorf(xs0), f1 = floorf(xs1), f2 = floorf(xs2);
        const float w0 = xs0 - f0, w1 = xs1 - f1, w2 = xs2 - f2;
        const unsigned int c0 = (unsigned int)f0;
        const unsigned int c1 = (unsigned int)f1;
        const unsigned int c2 = (unsigned int)f2;
        // spatial hash components (primes 1, 2654435761, 805459861)
        const unsigned int a0 = c0,                 a1 = c0 + 1u;
        const unsigned int b0 = c1 * 2654435761u,   b1 = b0 + 2654435761u;
        const unsigned int d0 = c2 * 805459861u,    d1 = d0 + 805459861u;

        const vf2* tb = (const vf2*)table + ((size_t)lvl << LOG2T);
        // 8 independent L2-resident gathers; issue all before consuming.
        vf2 g000 = tb[(a0 ^ b0 ^ d0) & HMASK];
        vf2 g001 = tb[(a0 ^ b0 ^ d1) & HMASK];
        vf2 g010 = tb[(a0 ^ b1 ^ d0) & HMASK];
        vf2 g011 = tb[(a0 ^ b1 ^ d1) & HMASK];
        vf2 g100 = tb[(a1 ^ b0 ^ d0) & HMASK];
        vf2 g101 = tb[(a1 ^ b0 ^ d1) & HMASK];
        vf2 g110 = tb[(a1 ^ b1 ^ d0) & HMASK];
        vf2 g111 = tb[(a1 ^ b1 ^ d1) & HMASK];

        const float u0 = 1.f - w0, v0 = 1.f - w1, s0 = 1.f - w2;
        const float m00 = v0 * s0, m01 = v0 * w2, m10 = w1 * s0, m11 = w1 * w2;
        const float p000 = u0 * m00, p001 = u0 * m01, p010 = u0 * m10, p011 = u0 * m11;
        const float p100 = w0 * m00, p101 = w0 * m01, p110 = w0 * m10, p111 = w0 * m11;

        float o0 =        p000 * g000.x;        float o1 =        p000 * g000.y;
        o0 = fmaf(p001, g001.x, o0);            o1 = fmaf(p001, g001.y, o1);
        o0 = fmaf(p010, g010.x, o0);            o1 = fmaf(p010, g010.y, o1);
        o0 = fmaf(p011, g011.x, o0);            o1 = fmaf(p011, g011.y, o1);
        o0 = fmaf(p100, g100.x, o0);            o1 = fmaf(p100, g100.y, o1);
        o0 = fmaf(p101, g101.x, o0);            o1 = fmaf(p101, g101.y, o1);
        o0 = fmaf(p110, g110.x, o0);            o1 = fmaf(p110, g110.y, o1);
        o0 = fmaf(p111, g111.x, o0);            o1 = fmaf(p111, g111.y, o1);

        // Non-temporal store: keep the 64MB table resident in the 192MB L2,
        // don't let the 128MB output stream thrash it. All 16 stores of this
        // thread hit one 128B line -> L2 write-combines to full-line HBM writes.
        vf2 rv = { o0, o1 };
        __builtin_nontemporal_store(rv, (vf2*)(orow + (lvl << 1)));
    }
}

extern "C" void kernel_launch(void* const* d_in, const int* in_sizes, int n_in,
                              void* d_out, int out_size, void* d_ws, size_t ws_size,
                              hipStream_t stream) {
    (void)n_in; (void)out_size; (void)d_ws; (void)ws_size;
    const float* x     = (const float*)d_in[0];
    const float* table = (const float*)d_in[1];
    float* out = (float*)d_out;
    const int npts = in_sizes[0] / 3;

    // Reproduce the reference's resolution computation bit-exactly in double
    // precision (CPython float ops == IEEE754 double via the same libm chain),
    // so borderline floors (e.g. level 15: 16*B^15 ~ 2048 +/- ulps) match.
    Res16 rs;
    const double B = std::exp((std::log(2048.0) - std::log(16.0)) / 15.0);
    for (int i = 0; i < LVLS; ++i)
        rs.v[i] = (float)std::floor(16.0 * std::pow(B, (double)i));

    const int blocks = (npts + 255) / 256;
    hipLaunchKernelGGL(hashgrid_fwd, dim3(blocks), dim3(256), 0, stream,
                       x, table, out, rs, npts);
}